// StrassenMHA_15272903704954
// MI455X (gfx1250) — compile-verified
//
#include <hip/hip_runtime.h>
#include <stdint.h>

// ---------------------------------------------------------------------------
// Strassen attention for MI455X (gfx1250), full-f32 WMMA pipeline.
// exp(clip(sim,±40)) reaches e^40 ≈ 2.4e17 -> fp16 overflows, bf16 risks
// cancellation error in num; V_WMMA_F32_16X16X4_F32 keeps exact f32 math.
// Total work ~36 GFLOP, working set ~19 MB (L2-resident at 192 MB).
// ---------------------------------------------------------------------------

typedef float v2f __attribute__((ext_vector_type(2)));
typedef float v8f __attribute__((ext_vector_type(8)));

constexpr int Bn   = 2;
constexpr int Nn   = 256;
constexpr int DIMn = 512;
constexpr int Hn   = 8;
constexpr int DHn  = 64;
constexpr int BH   = Bn * Hn;          // 16
constexpr int QKV_COLS = 5 * Hn * DHn; // 2560
constexpr float SCALE  = 0.125f;       // DH^-0.5
constexpr float CLAMPV = 40.0f;
constexpr float EPSF   = 1.1920929e-7f; // jnp.finfo(float32).eps

// Workspace layout (floats)
constexpr int SZ_T    = BH * Nn * DHn;     // 262144 per tensor (q,k1,k2,v1,v2)
constexpr int OFF_WS5 = 0;                 // 5 * SZ_T
constexpr int OFF_E1  = 5 * SZ_T;          // BH*N*N
constexpr int OFF_E2  = OFF_E1 + BH * Nn * Nn;
constexpr int OFF_E3T = OFF_E2 + BH * Nn * Nn;
constexpr int OFF_ATT = OFF_E3T + BH * Nn * Nn; // B*N*DIM

__device__ __forceinline__ v8f wmma4(v2f a, v2f b, v8f c) {
  // D = A(16x4 f32) * B(4x16 f32) + C(16x16 f32)  -> v_wmma_f32_16x16x4_f32
  return __builtin_amdgcn_wmma_f32_16x16x4_f32(
      /*neg_a=*/false, a, /*neg_b=*/false, b,
      /*c_mod=*/(short)0, c, /*reuse_a=*/false, /*reuse_b=*/false);
}

// ---------------------------------------------------------------------------
// K1: qkv = x @ w_qkv, scatter into split-head [t][b,h,n,d] layout.
// Grid: 8 (M/64) * 40 (Ncol/64) blocks, 128 threads (4 waves).
// ---------------------------------------------------------------------------
__global__ __launch_bounds__(128) void k_qkv(const float* __restrict__ x,
                                             const float* __restrict__ wqkv,
                                             float* __restrict__ ws5) {
  const int lane  = threadIdx.x & 31;
  const int wave  = threadIdx.x >> 5;
  const int lrow  = lane & 15;
  const int khalf = (lane >> 4) << 1;  // 0 or 2
  const int mhi   = (lane >> 4) << 3;  // 0 or 8
  const int nblk  = blockIdx.x % 40;
  const int mblk  = blockIdx.x / 40;
  const int mbase = mblk * 64 + wave * 16;
  const int nbase = nblk * 64;

  v8f acc[4] = {};
  const float* xrow = x + (mbase + lrow) * DIMn;
  for (int ks = 0; ks < DIMn / 4; ++ks) {
    const int k0 = ks * 4 + khalf;
    v2f a = *(const v2f*)(xrow + k0);
#pragma unroll
    for (int nt = 0; nt < 4; ++nt) {
      const int col = nbase + nt * 16 + lrow;
      v2f b;
      b.x = wqkv[k0 * QKV_COLS + col];
      b.y = wqkv[(k0 + 1) * QKV_COLS + col];
      acc[nt] = wmma4(a, b, acc[nt]);
    }
  }
#pragma unroll
  for (int nt = 0; nt < 4; ++nt) {
#pragma unroll
    for (int v = 0; v < 8; ++v) {
      const int row = mbase + v + mhi;          // global row in [0,512): b*256+n
      const int col = nbase + nt * 16 + lrow;   // [0,2560)
      const int t  = col >> 9;                  // tensor 0..4
      const int h  = (col >> 6) & 7;
      const int d  = col & 63;
      const int bb = row >> 8;
      const int n  = row & 255;
      ws5[t * SZ_T + ((bb * Hn + h) * Nn + n) * DHn + d] = acc[nt][v];
    }
  }
}

// ---------------------------------------------------------------------------
// K2: rmsnorm q/k1/k2 in place. One 64-wide row per wave (2 elems/lane).
// Grid: 3*BH*N / 8 = 1536 blocks, 256 threads (8 waves).
// ---------------------------------------------------------------------------
__global__ __launch_bounds__(256) void k_rmsnorm(float* __restrict__ ws5,
                                                 const float* __restrict__ qw,
                                                 const float* __restrict__ k1w,
                                                 const float* __restrict__ k2w) {
  const int lane = threadIdx.x & 31;
  const int wave = threadIdx.x >> 5;
  const int row  = blockIdx.x * 8 + wave;  // 0..12287
  const int t    = row >> 12;              // tensor 0..2 (4096 rows each)
  const int r    = row & 4095;
  float* p = ws5 + t * SZ_T + r * DHn;
  const float* w = (t == 0) ? qw : (t == 1) ? k1w : k2w;

  const float x0 = p[lane];
  const float x1 = p[lane + 32];
  float ss = x0 * x0 + x1 * x1;
#pragma unroll
  for (int off = 16; off > 0; off >>= 1) ss += __shfl_xor(ss, off, 32);
  const float rr = rsqrtf(ss * (1.0f / 64.0f) + EPSF);
  p[lane]      = x0 * rr * w[lane];
  p[lane + 32] = x1 * rr * w[lane + 32];
}

// ---------------------------------------------------------------------------
// K3: e1 = exp(clip(scale*q.k1^T)), e2 = exp(clip(scale*q.k2^T)),
//     e3t = exp(clip(scale*k2.k1^T))  (== e3 transposed, [k][j]).
// Grid: 3 * 16 (bh) * 16 (64x64 tiles) = 768 blocks, 128 threads.
// ---------------------------------------------------------------------------
__global__ __launch_bounds__(128) void k_sims(const float* __restrict__ ws5,
                                              float* __restrict__ e1,
                                              float* __restrict__ e2,
                                              float* __restrict__ e3t) {
  const int lane  = threadIdx.x & 31;
  const int wave  = threadIdx.x >> 5;
  const int lrow  = lane & 15;
  const int khalf = (lane >> 4) << 1;
  const int mhi   = (lane >> 4) << 3;

  int idx = blockIdx.x;
  const int tile  = idx & 15;  idx >>= 4;
  const int bh    = idx & 15;  idx >>= 4;
  const int which = idx;       // 0,1,2

  const float* q  = ws5 + 0 * SZ_T + bh * Nn * DHn;
  const float* k1 = ws5 + 1 * SZ_T + bh * Nn * DHn;
  const float* k2 = ws5 + 2 * SZ_T + bh * Nn * DHn;
  const float* L  = (which == 2) ? k2 : q;
  const float* R  = (which == 1) ? k2 : k1;
  float* out = ((which == 0) ? e1 : (which == 1) ? e2 : e3t) + bh * Nn * Nn;

  const int tm = (tile >> 2) * 64 + wave * 16;
  const int tn = (tile & 3) * 64;

  v8f acc[4] = {};
  const float* lr = L + (tm + lrow) * DHn;
  for (int ks = 0; ks < DHn / 4; ++ks) {
    const int k0 = ks * 4 + khalf;
    v2f a = *(const v2f*)(lr + k0);
#pragma unroll
    for (int nt = 0; nt < 4; ++nt) {
      // B[k][n] = R[n][k] -> contiguous v2f along k
      v2f b = *(const v2f*)(R + (tn + nt * 16 + lrow) * DHn + k0);
      acc[nt] = wmma4(a, b, acc[nt]);
    }
  }
#pragma unroll
  for (int nt = 0; nt < 4; ++nt) {
#pragma unroll
    for (int v = 0; v < 8; ++v) {
      const int row = tm + v + mhi;
      const int col = tn + nt * 16 + lrow;
      float s = acc[nt][v] * SCALE;
      s = fminf(fmaxf(s, -CLAMPV), CLAMPV);
      out[row * Nn + col] = expf(s);
    }
  }
}

// ---------------------------------------------------------------------------
// K4: core trilinear contraction, one WG per (bh, 16-row i-block).
//   per i:  T[j,d] = sum_k e2[i,k]*e3[j,k]*v2[k,d]   (M=256,K=256,N=64+den)
//   out[i,d] = sum_j e1[i,j]*v1[j,d]*T[j,d] / den[i]
// den is a virtual 5th d-tile (B-fragment == ones column). Two i's are
// processed per pass: the e3 A-fragment and v2 B-fragments are i-invariant,
// so this halves L2 traffic per WMMA. v2 is staged into LDS with
// GLOBAL_LOAD_ASYNC_TO_LDS_B128 (ASYNCcnt path, no VGPR round-trip).
// Grid: 16 (bh) * 16 (i-blocks) = 256 blocks, 256 threads (8 waves).
// ---------------------------------------------------------------------------
__global__ __launch_bounds__(256) void k_strassen(const float* __restrict__ ws5,
                                                  const float* __restrict__ e1,
                                                  const float* __restrict__ e2,
                                                  const float* __restrict__ e3t,
                                                  float* __restrict__ attn) {
  __shared__ __align__(16) float sV2[Nn * DHn];  // [k][d], 64 KB
  __shared__ float sE1[2][Nn];
  __shared__ float sE2[2][Nn];
  __shared__ float sOut[2][8][80];
  __shared__ float sTot[2][80];

  const int tid   = threadIdx.x;
  const int lane  = tid & 31;
  const int wave  = tid >> 5;
  const int lrow  = lane & 15;
  const int khalf = (lane >> 4) << 1;
  const int mhi   = (lane >> 4) << 3;

  const int bh   = blockIdx.x >> 4;
  const int iblk = blockIdx.x & 15;
  const int b    = bh >> 3;
  const int h    = bh & 7;

  const float* e1p = e1 + bh * Nn * Nn;
  const float* e2p = e2 + bh * Nn * Nn;
  const float* e3p = e3t + bh * Nn * Nn;  // [k][j]
  const float* v1p = ws5 + 3 * SZ_T + bh * Nn * DHn;
  const float* v2p = ws5 + 4 * SZ_T + bh * Nn * DHn;

  // Async-stage v2 (256x64 f32 = 64 KB) into LDS: 16 B per lane per issue.
  {
    const uint32_t lds0 = (uint32_t)(uintptr_t)(void*)sV2;  // addr[31:0] = LDS offset
    const uint64_t g0   = (uint64_t)(uintptr_t)(const void*)v2p;
    for (int t = tid; t < (Nn * DHn) / 4; t += 256) {
      uint32_t ldsa = lds0 + (uint32_t)(t * 16);
      uint64_t ga   = g0 + (uint64_t)(t * 16);
      asm volatile("global_load_async_to_lds_b128 %0, %1, off"
                   :: "v"(ldsa), "v"(ga) : "memory");
    }
    asm volatile("s_wait_asynccnt 0x0" ::: "memory");
  }
  __syncthreads();

  const int jt0 = wave * 2;  // this wave owns j-tiles jt0, jt0+1

  for (int it = 0; it < 8; ++it) {
    const int ig0 = iblk * 16 + it * 2;
    sE1[0][tid] = e1p[ig0 * Nn + tid];
    sE1[1][tid] = e1p[(ig0 + 1) * Nn + tid];
    sE2[0][tid] = e2p[ig0 * Nn + tid];
    sE2[1][tid] = e2p[(ig0 + 1) * Nn + tid];
    __syncthreads();

    v8f acc[2][2][5] = {};  // [i][jj][dt]
    for (int ks = 0; ks < Nn / 4; ++ks) {
      const int k0 = ks * 4 + khalf;
      v2f bf[5];
#pragma unroll
      for (int dt = 0; dt < 4; ++dt) {
        bf[dt].x = sV2[k0 * DHn + dt * 16 + lrow];
        bf[dt].y = sV2[(k0 + 1) * DHn + dt * 16 + lrow];
      }
      const float one = (lrow == 0) ? 1.0f : 0.0f;  // den column
      bf[4].x = one;
      bf[4].y = one;
      const float s00 = sE2[0][k0], s01 = sE2[0][k0 + 1];
      const float s10 = sE2[1][k0], s11 = sE2[1][k0 + 1];
#pragma unroll
      for (int jj = 0; jj < 2; ++jj) {
        const int j = (jt0 + jj) * 16 + lrow;
        const float g0 = e3p[k0 * Nn + j];        // e3[j,k0]
        const float g1 = e3p[(k0 + 1) * Nn + j];  // e3[j,k0+1]
        v2f af0, af1;
        af0.x = g0 * s00; af0.y = g1 * s01;
        af1.x = g0 * s10; af1.y = g1 * s11;
#pragma unroll
        for (int dt = 0; dt < 5; ++dt) {
          acc[0][jj][dt] = wmma4(af0, bf[dt], acc[0][jj][dt]);
          acc[1][jj][dt] = wmma4(af1, bf[dt], acc[1][jj][dt]);
        }
      }
    }

    // Epilogue: out[i,d] = sum_j e1[i,j]*v1[j,d]*T[j,d] (den tile: v1 == 1).
#pragma unroll
    for (int dt = 0; dt < 5; ++dt) {
      float sum0 = 0.0f, sum1 = 0.0f;
#pragma unroll
      for (int jj = 0; jj < 2; ++jj) {
#pragma unroll
        for (int v = 0; v < 8; ++v) {
          const int j = (jt0 + jj) * 16 + v + mhi;  // C-layout: M = v + 8*(lane>>4)
          const float w1 = (dt < 4) ? v1p[j * DHn + dt * 16 + lrow] : 1.0f;
          sum0 += sE1[0][j] * w1 * acc[0][jj][dt][v];
          sum1 += sE1[1][j] * w1 * acc[1][jj][dt][v];
        }
      }
      sum0 += __shfl_xor(sum0, 16, 32);  // combine lane halves (same d, j+8)
      sum1 += __shfl_xor(sum1, 16, 32);
      if (lane < 16) {
        sOut[0][wave][dt * 16 + lane] = sum0;
        sOut[1][wave][dt * 16 + lane] = sum1;
      }
    }
    __syncthreads();

    if (tid < 160) {
      const int i = tid / 80, c = tid % 80;
      float s = 0.0f;
#pragma unroll
      for (int w = 0; w < 8; ++w) s += sOut[i][w][c];
      sTot[i][c] = s;
    }
    __syncthreads();

    if (tid < 128) {
      const int i = tid >> 6, d = tid & 63;
      attn[(b * Nn + ig0 + i) * DIMn + h * DHn + d] = sTot[i][d] / sTot[i][64];
    }
    __syncthreads();  // protect sE1/sE2/sOut for next pass
  }
}

// ---------------------------------------------------------------------------
// K5: out = attn[512x512] @ w_out[512x512].
// Grid: 8*8 = 64 blocks, 128 threads (4 waves).
// ---------------------------------------------------------------------------
__global__ __launch_bounds__(128) void k_out(const float* __restrict__ attn,
                                             const float* __restrict__ wout,
                                             float* __restrict__ out) {
  const int lane  = threadIdx.x & 31;
  const int wave  = threadIdx.x >> 5;
  const int lrow  = lane & 15;
  const int khalf = (lane >> 4) << 1;
  const int mhi   = (lane >> 4) << 3;
  const int nblk  = blockIdx.x & 7;
  const int mblk  = blockIdx.x >> 3;
  const int mbase = mblk * 64 + wave * 16;
  const int nbase = nblk * 64;

  v8f acc[4] = {};
  const float* arow = attn + (mbase + lrow) * DIMn;
  for (int ks = 0; ks < DIMn / 4; ++ks) {
    const int k0 = ks * 4 + khalf;
    v2f a = *(const v2f*)(arow + k0);
#pragma unroll
    for (int nt = 0; nt < 4; ++nt) {
      const int col = nbase + nt * 16 + lrow;
      v2f b;
      b.x = wout[k0 * DIMn + col];
      b.y = wout[(k0 + 1) * DIMn + col];
      acc[nt] = wmma4(a, b, acc[nt]);
    }
  }
#pragma unroll
  for (int nt = 0; nt < 4; ++nt) {
#pragma unroll
    for (int v = 0; v < 8; ++v) {
      out[(mbase + v + mhi) * DIMn + nbase + nt * 16 + lrow] = acc[nt][v];
    }
  }
}

// ---------------------------------------------------------------------------
extern "C" void kernel_launch(void* const* d_in, const int* in_sizes, int n_in,
                              void* d_out, int out_size, void* d_ws, size_t ws_size,
                              hipStream_t stream) {
  (void)in_sizes; (void)n_in; (void)out_size; (void)ws_size;
  const float* x    = (const float*)d_in[0];
  const float* wqkv = (const float*)d_in[1];
  const float* wout = (const float*)d_in[2];
  const float* qw   = (const float*)d_in[3];
  const float* k1w  = (const float*)d_in[4];
  const float* k2w  = (const float*)d_in[5];

  float* ws   = (float*)d_ws;
  float* ws5  = ws + OFF_WS5;  // q,k1,k2,v1,v2 [b,h,n,d]
  float* e1   = ws + OFF_E1;
  float* e2   = ws + OFF_E2;
  float* e3t  = ws + OFF_E3T;
  float* attn = ws + OFF_ATT;  // [b,n,h*d]

  k_qkv<<<320, 128, 0, stream>>>(x, wqkv, ws5);
  k_rmsnorm<<<1536, 256, 0, stream>>>(ws5, qw, k1w, k2w);
  k_sims<<<768, 128, 0, stream>>>(ws5, e1, e2, e3t);
  k_strassen<<<256, 256, 0, stream>>>(ws5, e1, e2, e3t, attn);
  k_out<<<64, 128, 0, stream>>>(attn, wout, (float*)d_out);
}